// PairSelfAttentionLayer_79035988180995
// MI455X (gfx1250) — compile-verified
//
#include <hip/hip_runtime.h>
#include <hip/hip_bf16.h>

#define B_ 16
#define L_ 256
#define D_ 128

typedef float v2f __attribute__((ext_vector_type(2)));
typedef float v8f __attribute__((ext_vector_type(8)));

// D = A(16x4) * B(4x16) + C, all f32, wave32.  8-arg VOP3P form:
// (neg_a, A, neg_b, B, c_mod, C, reuse_a, reuse_b)
__device__ __forceinline__ v8f wmma_f32_16x16x4(v2f a, v2f b, v8f c) {
  return __builtin_amdgcn_wmma_f32_16x16x4_f32(false, a, false, b, (short)0, c,
                                               false, false);
}

__device__ __forceinline__ float wave_reduce_sum(float x) {
#pragma unroll
  for (int off = 16; off > 0; off >>= 1) x += __shfl_xor(x, off, 32);
  return x;
}

// ---------------------------------------------------------------------------
// Kernel 1: q_ = relu(q @ Wq + bq).  256 blocks x 16 rows, 8 waves per block,
// each wave owns one 16-wide column tile.  Wq staged in LDS (L2-resident).
// ---------------------------------------------------------------------------
__global__ void __launch_bounds__(256)
qproj_gemm_relu(const float* __restrict__ q, const float* __restrict__ Wq,
                const float* __restrict__ bq, float* __restrict__ q_act) {
  extern __shared__ float smem[];
  float* x_lds = smem;              // 16 x 128
  float* w_lds = smem + 16 * D_;    // 128 x 128
  const int row0 = blockIdx.x * 16;
  const int tid = threadIdx.x, lane = tid & 31, wv = tid >> 5;

  for (int idx = tid; idx < 16 * D_; idx += 256)
    x_lds[idx] = q[(size_t)row0 * D_ + idx];
  for (int idx = tid; idx < D_ * D_; idx += 256) w_lds[idx] = Wq[idx];
  __syncthreads();

  const int e0 = wv * 16, n = lane & 15, hi = lane >> 4;
  const int ka = hi ? 2 : 0, kb = hi ? 3 : 1;
  v8f acc = {};
#pragma unroll 8
  for (int d0 = 0; d0 < D_; d0 += 4) {
    v2f af, bf;
    af.x = x_lds[n * D_ + d0 + ka];          // A: m=lane%16, K per layout
    af.y = x_lds[n * D_ + d0 + kb];
    bf.x = w_lds[(d0 + ka) * D_ + e0 + n];   // B[k][n] = Wq[d0+k][e0+n]
    bf.y = w_lds[(d0 + kb) * D_ + e0 + n];
    acc = wmma_f32_16x16x4(af, bf, acc);
  }
  const float bias = bq[e0 + n];
#pragma unroll
  for (int r = 0; r < 8; ++r) {
    const int m = r + (hi ? 8 : 0);
    q_act[(size_t)(row0 + m) * D_ + e0 + n] = fmaxf(acc[r] + bias, 0.f);
  }
}

// ---------------------------------------------------------------------------
// Kernel 2: fused attention, one pass over `a` (online softmax).
// One workgroup per (batch, 16-row i-tile): 16 waves / 512 threads.
// LDS: q-tile 8KB | S_qk 16KB | (k+p) 128KB | (v+p) 128KB | mask 1KB = 281KB.
// Phase 1: each wave computes one 16x16 tile of S_qk = q_ @ k_^T via WMMA.
// Phase 2: wave w owns row i0+w, streams a[b,i,:,:] once, online softmax,
//          acc += alpha * (v_ + a)   (== alphav + alphaa).
// ---------------------------------------------------------------------------
__global__ void __launch_bounds__(512)
attn_flash(const float* __restrict__ q_act, const float* __restrict__ k,
           const float* __restrict__ v, const float* __restrict__ p,
           const float* __restrict__ a, const int* __restrict__ mask,
           float* __restrict__ att_pre) {
  extern __shared__ float smem[];
  float* q_lds = smem;                       // 16*128
  float* s_lds = q_lds + 16 * D_;            // 16*256
  float* k_lds = s_lds + 16 * L_;            // 256*128
  float* v_lds = k_lds + L_ * D_;            // 256*128
  int* m_lds = (int*)(v_lds + L_ * D_);      // 256

  const int blk = blockIdx.x;                // 256 blocks
  const int b = blk >> 4;                    // 16 i-tiles per batch
  const int i0 = (blk & 15) * 16;
  const int tid = threadIdx.x, lane = tid & 31, wv = tid >> 5;

  // Stage q-tile, k_+p, v_+p, mask into LDS (coalesced float4).
  for (int idx = tid; idx < 16 * D_; idx += 512)
    q_lds[idx] = q_act[((size_t)b * L_ + i0) * D_ + idx];
  {
    const float4* k4 = (const float4*)(k + (size_t)b * L_ * D_);
    const float4* v4 = (const float4*)(v + (size_t)b * L_ * D_);
    const float4* p4 = (const float4*)(p + (size_t)b * L_ * D_);
    float4* kl = (float4*)k_lds;
    float4* vl = (float4*)v_lds;
    for (int idx = tid; idx < L_ * D_ / 4; idx += 512) {
      const float4 pp = p4[idx];
      const float4 kk = k4[idx];
      const float4 vx = v4[idx];
      kl[idx] = make_float4(kk.x + pp.x, kk.y + pp.y, kk.z + pp.z, kk.w + pp.w);
      vl[idx] = make_float4(vx.x + pp.x, vx.y + pp.y, vx.z + pp.z, vx.w + pp.w);
    }
  }
  for (int idx = tid; idx < L_; idx += 512) m_lds[idx] = mask[b * L_ + idx];
  __syncthreads();

  // Phase 1: S_qk[0:16, 16w:16w+16] via WMMA, K=128 in steps of 4.
  {
    const int j0 = wv * 16, n = lane & 15, hi = lane >> 4;
    const int ka = hi ? 2 : 0, kb = hi ? 3 : 1;
    v8f acc = {};
#pragma unroll 8
    for (int d0 = 0; d0 < D_; d0 += 4) {
      v2f af, bf;
      af.x = q_lds[n * D_ + d0 + ka];
      af.y = q_lds[n * D_ + d0 + kb];
      bf.x = k_lds[(j0 + n) * D_ + d0 + ka];  // B[k][n] = k_[j0+n][d0+k]
      bf.y = k_lds[(j0 + n) * D_ + d0 + kb];
      acc = wmma_f32_16x16x4(af, bf, acc);
    }
#pragma unroll
    for (int r = 0; r < 8; ++r) {
      const int m = r + (hi ? 8 : 0);
      s_lds[m * L_ + j0 + n] = acc[r];
    }
  }
  __syncthreads();

  // Phase 2: wave wv owns row i = i0+wv.  Single streaming pass over a-row.
  const int i = i0 + wv;
  float qreg[4];
#pragma unroll
  for (int t = 0; t < 4; ++t) qreg[t] = q_lds[wv * D_ + lane * 4 + t];

  const float4* arow =
      (const float4*)a + ((size_t)(b * L_ + i)) * (L_ * D_ / 4);
  const float4* vl = (const float4*)v_lds;
  const float inv_sqrt_d = 0.08838834764831845f;  // 1/sqrt(128)

  float mrun = -1e30f, lrun = 0.f;
  float acc0 = 0.f, acc1 = 0.f, acc2 = 0.f, acc3 = 0.f;

  for (int j = 0; j < L_; ++j) {
    const float4 av = arow[j * 32 + lane];  // 512B/wave, fully coalesced
    if (j + 8 < L_)
      __builtin_prefetch(arow + (j + 8) * 32 + lane, 0, 0);
    float part = qreg[0] * av.x + qreg[1] * av.y + qreg[2] * av.z +
                 qreg[3] * av.w;
    part = wave_reduce_sum(part);           // qa[i,j], all lanes
    float s = (s_lds[wv * L_ + j] + part) * inv_sqrt_d;
    if (m_lds[j] == 0) s = -1e12f;
    const float mnew = fmaxf(mrun, s);
    const float scale = __expf(mrun - mnew);
    const float pw = __expf(s - mnew);
    lrun = lrun * scale + pw;
    const float4 vv = vl[j * 32 + lane];
    acc0 = acc0 * scale + pw * (vv.x + av.x);
    acc1 = acc1 * scale + pw * (vv.y + av.y);
    acc2 = acc2 * scale + pw * (vv.z + av.z);
    acc3 = acc3 * scale + pw * (vv.w + av.w);
    mrun = mnew;
  }
  const float invl = 1.f / lrun;
  float4 outv = make_float4(acc0 * invl, acc1 * invl, acc2 * invl, acc3 * invl);
  ((float4*)(att_pre + ((size_t)(b * L_ + i)) * D_))[lane] = outv;
}

// ---------------------------------------------------------------------------
// Kernel 3: h = relu(att@W1+b1); out = h@W2+b2+att; LayerNorm; scatter to
// d_out (c region then att_v region).  W-region of LDS reused for W1 then W2.
// ---------------------------------------------------------------------------
__global__ void __launch_bounds__(256)
ffn_ln(const float* __restrict__ att_pre, const float* __restrict__ W1,
       const float* __restrict__ b1, const float* __restrict__ W2,
       const float* __restrict__ b2, const float* __restrict__ ln_g,
       const float* __restrict__ ln_b, float* __restrict__ out) {
  extern __shared__ float smem[];
  float* x_lds = smem;                   // 16*128 (att, kept for residual)
  float* w_lds = x_lds + 16 * D_;        // 128*128 (W1, then W2)
  float* h_lds = w_lds + D_ * D_;        // 16*128
  float* o_lds = h_lds + 16 * D_;        // 16*128

  const int row0 = blockIdx.x * 16;
  const int tid = threadIdx.x, lane = tid & 31, wv = tid >> 5;
  const int e0 = wv * 16, n = lane & 15, hi = lane >> 4;
  const int ka = hi ? 2 : 0, kb = hi ? 3 : 1;

  for (int idx = tid; idx < 16 * D_; idx += 256)
    x_lds[idx] = att_pre[(size_t)row0 * D_ + idx];
  for (int idx = tid; idx < D_ * D_; idx += 256) w_lds[idx] = W1[idx];
  __syncthreads();

  {  // GEMM 1 + relu -> h_lds
    v8f acc = {};
#pragma unroll 8
    for (int d0 = 0; d0 < D_; d0 += 4) {
      v2f af, bf;
      af.x = x_lds[n * D_ + d0 + ka];
      af.y = x_lds[n * D_ + d0 + kb];
      bf.x = w_lds[(d0 + ka) * D_ + e0 + n];
      bf.y = w_lds[(d0 + kb) * D_ + e0 + n];
      acc = wmma_f32_16x16x4(af, bf, acc);
    }
    const float bias = b1[e0 + n];
#pragma unroll
    for (int r = 0; r < 8; ++r) {
      const int m = r + (hi ? 8 : 0);
      h_lds[m * D_ + e0 + n] = fmaxf(acc[r] + bias, 0.f);
    }
  }
  __syncthreads();
  for (int idx = tid; idx < D_ * D_; idx += 256) w_lds[idx] = W2[idx];
  __syncthreads();

  {  // GEMM 2 + bias + residual -> o_lds
    v8f acc = {};
#pragma unroll 8
    for (int d0 = 0; d0 < D_; d0 += 4) {
      v2f af, bf;
      af.x = h_lds[n * D_ + d0 + ka];
      af.y = h_lds[n * D_ + d0 + kb];
      bf.x = w_lds[(d0 + ka) * D_ + e0 + n];
      bf.y = w_lds[(d0 + kb) * D_ + e0 + n];
      acc = wmma_f32_16x16x4(af, bf, acc);
    }
    const float bias = b2[e0 + n];
#pragma unroll
    for (int r = 0; r < 8; ++r) {
      const int m = r + (hi ? 8 : 0);
      o_lds[m * D_ + e0 + n] = acc[r] + bias + x_lds[m * D_ + e0 + n];
    }
  }
  __syncthreads();

  // LayerNorm: 8 waves x 2 rows; lane covers 4 columns.
  const float4 g4 = ((const float4*)ln_g)[lane];
  const float4 be4 = ((const float4*)ln_b)[lane];
#pragma unroll
  for (int rr = 0; rr < 2; ++rr) {
    const int m = wv * 2 + rr;
    const float4 xv = ((const float4*)(o_lds + m * D_))[lane];
    const float sum = wave_reduce_sum(xv.x + xv.y + xv.z + xv.w);
    const float sq = wave_reduce_sum(xv.x * xv.x + xv.y * xv.y +
                                     xv.z * xv.z + xv.w * xv.w);
    const float mu = sum * (1.f / D_);
    const float var = sq * (1.f / D_) - mu * mu;
    const float rstd = rsqrtf(var + 1e-5f);
    float4 y;
    y.x = (xv.x - mu) * rstd * g4.x + be4.x;
    y.y = (xv.y - mu) * rstd * g4.y + be4.y;
    y.z = (xv.z - mu) * rstd * g4.z + be4.z;
    y.w = (xv.w - mu) * rstd * g4.w + be4.w;
    const int gr = row0 + m;           // flat (b*L + l)
    const int bb = gr >> 8, ll = gr & (L_ - 1);
    ((float4*)(out + (size_t)B_ * D_ + (size_t)gr * D_))[lane] = y;  // att_v
    if (ll == L_ - 1)
      ((float4*)(out + (size_t)bb * D_))[lane] = y;                  // c
  }
}

// ---------------------------------------------------------------------------
extern "C" void kernel_launch(void* const* d_in, const int* in_sizes, int n_in,
                              void* d_out, int out_size, void* d_ws,
                              size_t ws_size, hipStream_t stream) {
  (void)in_sizes; (void)n_in; (void)out_size; (void)ws_size;
  const float* q = (const float*)d_in[0];
  const float* k = (const float*)d_in[1];
  const float* v = (const float*)d_in[2];
  const float* p = (const float*)d_in[3];
  const float* a = (const float*)d_in[4];
  const int* mask = (const int*)d_in[5];
  const float* Wq = (const float*)d_in[6];
  const float* bq = (const float*)d_in[7];
  const float* W1 = (const float*)d_in[8];
  const float* b1 = (const float*)d_in[9];
  const float* W2 = (const float*)d_in[10];
  const float* b2 = (const float*)d_in[11];
  const float* ln_g = (const float*)d_in[12];
  const float* ln_b = (const float*)d_in[13];
  float* out = (float*)d_out;

  float* q_act = (float*)d_ws;                       // 4096 x 128 f32 (2 MB)
  float* att_pre = q_act + (size_t)B_ * L_ * D_;     // 4096 x 128 f32 (2 MB)

  const size_t sm1 = (size_t)(16 * D_ + D_ * D_) * sizeof(float);
  const size_t sm2 =
      (size_t)(16 * D_ + 16 * L_ + 2 * L_ * D_) * sizeof(float) +
      (size_t)L_ * sizeof(int);
  const size_t sm3 = (size_t)(3 * 16 * D_ + D_ * D_) * sizeof(float);

  qproj_gemm_relu<<<B_ * L_ / 16, 256, sm1, stream>>>(q, Wq, bq, q_act);
  attn_flash<<<B_ * L_ / 16, 512, sm2, stream>>>(q_act, k, v, p, a, mask,
                                                 att_pre);
  ffn_ln<<<B_ * L_ / 16, 256, sm3, stream>>>(att_pre, W1, b1, W2, b2, ln_g,
                                             ln_b, out);
}